// DPSViTModel_10694468567193
// MI455X (gfx1250) — compile-verified
//
#include <hip/hip_runtime.h>
#include <hip/hip_bf16.h>

#define K_TOP       16
#define NUM_SAMPLES 500
#define SIGMA       0.05f
#define D_DIM       256   // hs*ws
#define B_DIM       8
#define C_DIM       3
#define HW          512
#define PATCH_PIX   4096  // 64*64
#define PAD_TOP     16
#define SCALE       32
#define PADH        544   // rows needed: max index 15*32+63 = 543
#define PADW        576   // stride: 2304 B = 18*128 B, rows 128B-aligned

typedef float v2f __attribute__((ext_vector_type(2)));
typedef float v8f __attribute__((ext_vector_type(8)));

// ---------------- workspace layout ----------------
// floats [0, 2048)             : s_norm  8 x 256
// floats [2048, 2048+32768)    : counts  8 x 16 x 256
// floats [40960, +24*544*576)  : xpad    (only if ws_size permits)
#define SNORM_OFF  0
#define COUNTS_OFF 2048
#define XPAD_OFF   40960   // float offset, 160 KB in, 256B-aligned
#define XPAD_ELEMS ((size_t)B_DIM * C_DIM * PADH * PADW)

__global__ void zero_counts_kernel(float* __restrict__ counts) {
  int t = blockIdx.x * blockDim.x + threadIdx.x;
  if (t < B_DIM * K_TOP * D_DIM) counts[t] = 0.0f;
}

// one block per batch, 256 threads; min/max over 256 scores, then normalize
__global__ void normalize_kernel(const float* __restrict__ scores,
                                 float* __restrict__ s_norm) {
  __shared__ float red_min[D_DIM];
  __shared__ float red_max[D_DIM];
  int b = blockIdx.x;
  int t = threadIdx.x;
  float v = scores[b * D_DIM + t];
  red_min[t] = v;
  red_max[t] = v;
  __syncthreads();
  for (int s = D_DIM / 2; s > 0; s >>= 1) {
    if (t < s) {
      red_min[t] = fminf(red_min[t], red_min[t + s]);
      red_max[t] = fmaxf(red_max[t], red_max[t + s]);
    }
    __syncthreads();
  }
  float mn = red_min[0], mx = red_max[0];
  s_norm[b * D_DIM + t] = (v - mn) / (mx - mn + 1e-5f);
}

// one thread per (batch, sample): top-16, sort indices, histogram.
__global__ void topk_hist_kernel(const float* __restrict__ s_norm,
                                 const float* __restrict__ noise,
                                 float* __restrict__ counts) {
  int gid = blockIdx.x * blockDim.x + threadIdx.x;
  if (gid >= B_DIM * NUM_SAMPLES) return;
  int b = gid / NUM_SAMPLES;
  const float* sb = s_norm + b * D_DIM;
  const float* nz = noise + (size_t)gid * D_DIM;

  float tv[K_TOP];
  int   ti[K_TOP];
#pragma unroll
  for (int k = 0; k < K_TOP; ++k) { tv[k] = -3.0e38f; ti[k] = 0; }

  for (int d = 0; d < D_DIM; ++d) {
    float p = sb[d] + nz[d] * SIGMA;
    if (p > tv[K_TOP - 1]) {
      int pos = K_TOP - 1;
      while (pos > 0 && tv[pos - 1] < p) {
        tv[pos] = tv[pos - 1];
        ti[pos] = ti[pos - 1];
        --pos;
      }
      tv[pos] = p;
      ti[pos] = d;
    }
  }
  for (int a = 1; a < K_TOP; ++a) {
    int key = ti[a];
    int pos = a - 1;
    while (pos >= 0 && ti[pos] > key) { ti[pos + 1] = ti[pos]; --pos; }
    ti[pos + 1] = key;
  }
  const float inv = 1.0f / (float)NUM_SAMPLES;
#pragma unroll
  for (int k = 0; k < K_TOP; ++k)
    atomicAdd(&counts[(b * K_TOP + k) * D_DIM + ti[k]], inv);
}

// Materialize the zero-padded image: xpad[bc][r][c] (PADH x PADW, stride PADW)
// grid = (PADW/192, PADH, B*C), block = 192
__global__ void pad_kernel(const float* __restrict__ xh,
                           float* __restrict__ xpad) {
  int cc = blockIdx.x * blockDim.x + threadIdx.x;  // 0..575
  int r  = blockIdx.y;                             // 0..543
  int bc = blockIdx.z;                             // 0..23
  int sr = r - PAD_TOP;
  int sc = cc - PAD_TOP;
  float v = 0.0f;
  if ((unsigned)sr < (unsigned)HW && (unsigned)sc < (unsigned)HW)
    v = xh[((size_t)bc * HW + sr) * HW + sc];
  xpad[((size_t)bc * PADH + r) * PADW + cc] = v;
}

// ---------- fast path: GEMM on pre-padded image (no bounds logic) ----------
// Per (b,c): out[16 x 4096] = indicators[16 x 256] * patches[256 x 4096]
// via V_WMMA_F32_16X16X4_F32.  One wave owns one 64-pixel patch row (ph
// uniform); one ds_load_b64 of A feeds 4 WMMAs.  grid=(8,3,8), block=256.
__global__ void einsum_wmma_pad_kernel(const float* __restrict__ xpad,
                                       const float* __restrict__ w,
                                       float* __restrict__ out) {
  __shared__ float lds_w[K_TOP * D_DIM];  // 16 KB
  int b = blockIdx.z;
  int c = blockIdx.y;

  for (int t = threadIdx.x; t < K_TOP * D_DIM; t += blockDim.x)
    lds_w[t] = w[b * K_TOP * D_DIM + t];
  __syncthreads();

  int wave = threadIdx.x >> 5;
  int lane = threadIdx.x & 31;
  int n    = lane & 15;
  int half = lane >> 4;

  int ph = blockIdx.x * 8 + wave;  // patch row 0..63, uniform per wave
  const float* xbc = xpad + ((size_t)(b * C_DIM + c)) * PADH * PADW;

  v8f acc0 = {}, acc1 = {}, acc2 = {}, acc3 = {};

  for (int ij0 = 0; ij0 < D_DIM; ij0 += 4) {
    v2f A;  // A (16x4): lanes 0-15 -> K0(V0),K1(V1); lanes 16-31 -> K2,K3
    A.x = lds_w[n * D_DIM + ij0 + half * 2 + 0];
    A.y = lds_w[n * D_DIM + ij0 + half * 2 + 1];

    int rx = ij0 + half;      // B row for B*.x
    int ry = ij0 + 2 + half;  // B row for B*.y
    const float* px = xbc + ((rx >> 4) * SCALE + ph) * PADW + (rx & 15) * SCALE + n;
    const float* py = xbc + ((ry >> 4) * SCALE + ph) * PADW + (ry & 15) * SCALE + n;

    v2f B0, B1, B2, B3;
    B0.x = px[0];  B0.y = py[0];
    B1.x = px[16]; B1.y = py[16];
    B2.x = px[32]; B2.y = py[32];
    B3.x = px[48]; B3.y = py[48];

    acc0 = __builtin_amdgcn_wmma_f32_16x16x4_f32(false, A, false, B0,
                                                 (short)0, acc0, false, false);
    acc1 = __builtin_amdgcn_wmma_f32_16x16x4_f32(false, A, false, B1,
                                                 (short)0, acc1, false, false);
    acc2 = __builtin_amdgcn_wmma_f32_16x16x4_f32(false, A, false, B2,
                                                 (short)0, acc2, false, false);
    acc3 = __builtin_amdgcn_wmma_f32_16x16x4_f32(false, A, false, B3,
                                                 (short)0, acc3, false, false);
  }

#pragma unroll
  for (int r = 0; r < 8; ++r) {
    int M = r + half * 8;
    size_t row = (((size_t)(b * K_TOP + M)) * C_DIM + c) * PATCH_PIX + ph * 64;
    __builtin_nontemporal_store(acc0[r], &out[row + n]);
    __builtin_nontemporal_store(acc1[r], &out[row + n + 16]);
    __builtin_nontemporal_store(acc2[r], &out[row + n + 32]);
    __builtin_nontemporal_store(acc3[r], &out[row + n + 48]);
  }
}

// ---------- fallback path: clamp + cndmask gather (no scratch image) -------
__device__ __forceinline__ float gather(const float* __restrict__ xbc,
                                        int ij, int ph, int pw) {
  int i  = ij >> 4;
  int j  = ij & 15;
  int r  = i * SCALE + ph - PAD_TOP;
  int cl = j * SCALE + pw - PAD_TOP;
  bool ok = ((unsigned)r < (unsigned)HW) & ((unsigned)cl < (unsigned)HW);
  int rc = min(max(r, 0), HW - 1);
  int cc = min(max(cl, 0), HW - 1);
  float v = xbc[rc * HW + cc];
  return ok ? v : 0.0f;
}

__global__ void einsum_wmma_clamp_kernel(const float* __restrict__ xh,
                                         const float* __restrict__ w,
                                         float* __restrict__ out) {
  __shared__ float lds_w[K_TOP * D_DIM];
  int b = blockIdx.z;
  int c = blockIdx.y;

  for (int t = threadIdx.x; t < K_TOP * D_DIM; t += blockDim.x)
    lds_w[t] = w[b * K_TOP * D_DIM + t];
  __syncthreads();

  int wave = threadIdx.x >> 5;
  int lane = threadIdx.x & 31;
  int n    = lane & 15;
  int half = lane >> 4;

  int ph = blockIdx.x * 8 + wave;
  const float* xbc = xh + ((size_t)b * C_DIM + c) * HW * HW;

  v8f acc0 = {}, acc1 = {}, acc2 = {}, acc3 = {};

  for (int ij0 = 0; ij0 < D_DIM; ij0 += 4) {
    v2f A;
    A.x = lds_w[n * D_DIM + ij0 + half * 2 + 0];
    A.y = lds_w[n * D_DIM + ij0 + half * 2 + 1];

    int rx = ij0 + half;
    int ry = ij0 + 2 + half;

    v2f B0, B1, B2, B3;
    B0.x = gather(xbc, rx, ph, n);       B0.y = gather(xbc, ry, ph, n);
    B1.x = gather(xbc, rx, ph, n + 16);  B1.y = gather(xbc, ry, ph, n + 16);
    B2.x = gather(xbc, rx, ph, n + 32);  B2.y = gather(xbc, ry, ph, n + 32);
    B3.x = gather(xbc, rx, ph, n + 48);  B3.y = gather(xbc, ry, ph, n + 48);

    acc0 = __builtin_amdgcn_wmma_f32_16x16x4_f32(false, A, false, B0,
                                                 (short)0, acc0, false, false);
    acc1 = __builtin_amdgcn_wmma_f32_16x16x4_f32(false, A, false, B1,
                                                 (short)0, acc1, false, false);
    acc2 = __builtin_amdgcn_wmma_f32_16x16x4_f32(false, A, false, B2,
                                                 (short)0, acc2, false, false);
    acc3 = __builtin_amdgcn_wmma_f32_16x16x4_f32(false, A, false, B3,
                                                 (short)0, acc3, false, false);
  }

#pragma unroll
  for (int r = 0; r < 8; ++r) {
    int M = r + half * 8;
    size_t row = (((size_t)(b * K_TOP + M)) * C_DIM + c) * PATCH_PIX + ph * 64;
    __builtin_nontemporal_store(acc0[r], &out[row + n]);
    __builtin_nontemporal_store(acc1[r], &out[row + n + 16]);
    __builtin_nontemporal_store(acc2[r], &out[row + n + 32]);
    __builtin_nontemporal_store(acc3[r], &out[row + n + 48]);
  }
}

extern "C" void kernel_launch(void* const* d_in, const int* in_sizes, int n_in,
                              void* d_out, int out_size, void* d_ws, size_t ws_size,
                              hipStream_t stream) {
  const float* x_high = (const float*)d_in[0];  // (8,3,512,512)
  const float* scores = (const float*)d_in[1];  // (8,16,16)
  const float* noise  = (const float*)d_in[2];  // (8,500,256)
  float* out = (float*)d_out;                   // (128,3,64,64)

  float* ws     = (float*)d_ws;
  float* s_norm = ws + SNORM_OFF;
  float* counts = ws + COUNTS_OFF;
  float* xpad   = ws + XPAD_OFF;

  zero_counts_kernel<<<(B_DIM * K_TOP * D_DIM + 255) / 256, 256, 0, stream>>>(counts);
  normalize_kernel<<<B_DIM, D_DIM, 0, stream>>>(scores, s_norm);
  topk_hist_kernel<<<(B_DIM * NUM_SAMPLES + 255) / 256, 256, 0, stream>>>(s_norm, noise, counts);

  dim3 grid(8, C_DIM, B_DIM);  // 8 blocks x 8 waves = 64 patch rows per (b,c)
  size_t need_bytes = ((size_t)XPAD_OFF + XPAD_ELEMS) * sizeof(float);
  if (ws_size >= need_bytes) {
    dim3 pgrid(PADW / 192, PADH, B_DIM * C_DIM);
    pad_kernel<<<pgrid, 192, 0, stream>>>(x_high, xpad);
    einsum_wmma_pad_kernel<<<grid, 256, 0, stream>>>(xpad, counts, out);
  } else {
    einsum_wmma_clamp_kernel<<<grid, 256, 0, stream>>>(x_high, counts, out);
  }
}